// ClimateRiskGNN_10368051052674
// MI455X (gfx1250) — compile-verified
//
#include <hip/hip_runtime.h>
#include <math.h>

#define NN 100000
#define NE 800000
#define ET (NE + NN)   // edges + self loops

typedef __attribute__((ext_vector_type(2))) float v2f;
typedef __attribute__((ext_vector_type(8))) float v8f;

// ---- order-preserving float <-> uint for atomicMax on floats ----
__device__ __forceinline__ unsigned f2ord(float f) {
  unsigned u = __float_as_uint(f);
  return (u & 0x80000000u) ? ~u : (u | 0x80000000u);
}
__device__ __forceinline__ float ord2f(unsigned k) {
  unsigned u = (k & 0x80000000u) ? (k & 0x7fffffffu) : ~k;
  return __uint_as_float(u);
}

// =====================================================================
// f32 WMMA GEMM: Out[M,Nc] = A[M,K] @ W[K,Nc] (+bias).
// One wave32 per (16-row) x (NT*16-col) block of output.
// V_WMMA_F32_16X16X4_F32 keeps full f32 precision; K templated so the
// K=64 path fully unrolls (no guards, b64 A loads) and the K=7 path
// uses clamped addresses + v_cndmask zeroing instead of exec branches.
// A frag (16x4): lane%16 = M; lanes 0-15 hold K = kk+{0,1}, lanes 16-31
// hold K = kk+{2,3}. B frag (4x16): same split over K rows, lane%16 = N.
// C/D: VGPR v -> M = v + (lane/16)*8, N = lane%16.
// =====================================================================
template <int K, int NT>
__global__ __launch_bounds__(32)
void wmma_gemm(const float* __restrict__ A, const float* __restrict__ W,
               const float* __restrict__ bias, float* __restrict__ Out,
               int Nc) {
  const unsigned lane = threadIdx.x;
  const unsigned half = lane >> 4;
  const unsigned l16  = lane & 15u;
  const unsigned row  = blockIdx.x * 16u + l16;
  const unsigned col0 = blockIdx.y * (16u * NT) + l16;
  v8f acc[NT];
#pragma unroll
  for (int t = 0; t < NT; ++t) acc[t] = (v8f){};

#pragma unroll
  for (int kk = 0; kk < K; kk += 4) {
    const unsigned k0 = kk + half * 2u;
    const unsigned k1 = k0 + 1u;
    v2f a;
    if constexpr ((K & 3) == 0) {
      a = *(const v2f*)(A + (size_t)row * K + k0);   // aligned b64 load
    } else {
      // clamp address (stays in-bounds), zero via select -> v_cndmask
      const unsigned ka = k0 < K ? k0 : K - 1;
      const unsigned kb = k1 < K ? k1 : K - 1;
      const float ax = A[row * K + ka];
      const float ay = A[row * K + kb];
      a.x = (k0 < K) ? ax : 0.0f;
      a.y = (k1 < K) ? ay : 0.0f;
    }
#pragma unroll
    for (int t = 0; t < NT; ++t) {
      const unsigned c = col0 + t * 16u;
      v2f b;
      if constexpr ((K & 3) == 0) {
        b.x = W[k0 * Nc + c];
        b.y = W[k1 * Nc + c];
      } else {
        const unsigned ka = k0 < K ? k0 : K - 1;
        const unsigned kb = k1 < K ? k1 : K - 1;
        const float bx = W[ka * Nc + c];
        const float by = W[kb * Nc + c];
        b.x = (k0 < K) ? bx : 0.0f;
        b.y = (k1 < K) ? by : 0.0f;
      }
      acc[t] = __builtin_amdgcn_wmma_f32_16x16x4_f32(false, a, false, b,
                                                     (short)0, acc[t],
                                                     false, false);
    }
  }

#pragma unroll
  for (int t = 0; t < NT; ++t) {
    const unsigned c = col0 + t * 16u;
    const float bv = bias ? bias[c] : 0.0f;
#pragma unroll
    for (int v = 0; v < 8; ++v) {
      const unsigned m = blockIdx.x * 16u + v + half * 8u;
      Out[(size_t)m * Nc + c] = acc[t][v] + bv;
    }
  }
}

// ---------------- small utility kernels ----------------
__global__ void fill_f32(float* p, float v, int n) {
  int i = blockIdx.x * blockDim.x + threadIdx.x;
  if (i < n) p[i] = v;
}
__global__ void fill_ord(unsigned* p, float v, int n) {
  int i = blockIdx.x * blockDim.x + threadIdx.x;
  if (i < n) p[i] = f2ord(v);
}
__global__ void deg_accum(const int* __restrict__ ei, float* deg) {
  int e = blockIdx.x * blockDim.x + threadIdx.x;
  if (e < NE) atomicAdd(&deg[ei[NE + e]], 1.0f);
}
__global__ void rsqrt_ip(float* p, int n) {
  int i = blockIdx.x * blockDim.x + threadIdx.x;
  if (i < n) p[i] = rsqrtf(p[i]);
}

// per-node per-head attention dots: a_s[n,h] = <hfeat[n,h,:], a_src[h,:]>
__global__ void att_dots(const float* __restrict__ hfeat,
                         const float* __restrict__ asrc,
                         const float* __restrict__ adst,
                         float* __restrict__ a_s, float* __restrict__ a_d,
                         int Hh) {
  int gid = blockIdx.x * blockDim.x + threadIdx.x;
  if (gid >= NN * Hh) return;
  int n = gid / Hh, h = gid % Hh;
  const float* hv = hfeat + (size_t)n * Hh * 64 + h * 64;
  float s = 0.f, d = 0.f;
#pragma unroll 8
  for (int c = 0; c < 64; ++c) {
    s += hv[c] * asrc[h * 64 + c];
    d += hv[c] * adst[h * 64 + c];
  }
  a_s[gid] = s;
  a_d[gid] = d;
}

__device__ __forceinline__ void edge_sd(const int* ei, int e, int& s, int& d) {
  if (e < NE) { s = ei[e]; d = ei[NE + e]; }
  else        { s = e - NE; d = e - NE; }
}
__device__ __forceinline__ float lrelu(float v) { return v > 0.f ? v : 0.2f * v; }

__global__ void gat_edge_max(const int* __restrict__ ei,
                             const float* __restrict__ a_s,
                             const float* __restrict__ a_d,
                             unsigned* __restrict__ emax, int Hh) {
  int gid = blockIdx.x * blockDim.x + threadIdx.x;
  if (gid >= ET * Hh) return;
  int e = gid / Hh, h = gid % Hh, s, d;
  edge_sd(ei, e, s, d);
  float v = lrelu(a_s[s * Hh + h] + a_d[d * Hh + h]);
  atomicMax(&emax[d * Hh + h], f2ord(v));
}

__global__ void gat_edge_expsum(const int* __restrict__ ei,
                                const float* __restrict__ a_s,
                                const float* __restrict__ a_d,
                                const unsigned* __restrict__ emax,
                                float* __restrict__ denom, int Hh) {
  int gid = blockIdx.x * blockDim.x + threadIdx.x;
  if (gid >= ET * Hh) return;
  int e = gid / Hh, h = gid % Hh, s, d;
  edge_sd(ei, e, s, d);
  float v = lrelu(a_s[s * Hh + h] + a_d[d * Hh + h]);
  atomicAdd(&denom[d * Hh + h], __expf(v - ord2f(emax[d * Hh + h])));
}

__global__ void gat_alpha(const int* __restrict__ ei,
                          const float* __restrict__ a_s,
                          const float* __restrict__ a_d,
                          const unsigned* __restrict__ emax,
                          const float* __restrict__ denom,
                          float* __restrict__ alpha, int Hh) {
  int gid = blockIdx.x * blockDim.x + threadIdx.x;
  if (gid >= ET * Hh) return;
  int e = gid / Hh, h = gid % Hh, s, d;
  edge_sd(ei, e, s, d);
  float v = lrelu(a_s[s * Hh + h] + a_d[d * Hh + h]);
  alpha[gid] = __expf(v - ord2f(emax[d * Hh + h])) / (denom[d * Hh + h] + 1e-16f);
}

// weighted aggregation: agg[dst, j] += hfeat[src, j] * alpha[e, j/64]
__global__ void gat_edge_agg(const int* __restrict__ ei,
                             const float* __restrict__ hfeat,
                             const float* __restrict__ alpha,
                             float* __restrict__ agg, int Hh, int cols) {
  long long gid = (long long)blockIdx.x * blockDim.x + threadIdx.x;
  if (gid >= (long long)ET * cols) return;
  int e = (int)(gid / cols), j = (int)(gid % cols), s, d;
  edge_sd(ei, e, s, d);
  int h = j >> 6;
  atomicAdd(&agg[(size_t)d * cols + j],
            hfeat[(size_t)s * cols + j] * alpha[e * Hh + h]);
}

// gcn aggregation: agg[dst,c] += hfeat[src,c] * dinv[src]*dinv[dst]
__global__ void gcn_edge_agg(const int* __restrict__ ei,
                             const float* __restrict__ hfeat,
                             const float* __restrict__ dinv,
                             float* __restrict__ agg) {
  long long gid = (long long)blockIdx.x * blockDim.x + threadIdx.x;
  if (gid >= (long long)ET * 64) return;
  int e = (int)(gid >> 6), c = (int)(gid & 63), s, d;
  edge_sd(ei, e, s, d);
  atomicAdd(&agg[(size_t)d * 64 + c],
            hfeat[(size_t)s * 64 + c] * dinv[s] * dinv[d]);
}

// head-mean + bias + eval-BatchNorm + ELU + residual  -> out
__global__ void post_kernel(const float* __restrict__ agg, int heads,
                            const float* __restrict__ b,
                            const float* __restrict__ gamma,
                            const float* __restrict__ beta,
                            const float* __restrict__ mean,
                            const float* __restrict__ var,
                            const float* __restrict__ res,
                            float* __restrict__ out) {
  int gid = blockIdx.x * blockDim.x + threadIdx.x;
  if (gid >= NN * 64) return;
  int n = gid >> 6, c = gid & 63;
  float s = 0.f;
  for (int h = 0; h < heads; ++h) s += agg[(size_t)n * heads * 64 + h * 64 + c];
  s = s / (float)heads + b[c];
  s = (s - mean[c]) * rsqrtf(var[c] + 1e-5f) * gamma[c] + beta[c];
  s = s > 0.f ? s : (__expf(s) - 1.f);  // ELU
  out[gid] = s + res[gid];
}

// fused MLP head: 64->32->16->1, relu/relu/sigmoid; weights in LDS
__global__ __launch_bounds__(256)
void head_kernel(const float* __restrict__ h, const float* __restrict__ W1,
                 const float* __restrict__ b1, const float* __restrict__ W2,
                 const float* __restrict__ b2, const float* __restrict__ W3,
                 const float* __restrict__ b3, float* __restrict__ out) {
  __shared__ float sW1[64 * 32], sW2[32 * 16], sb1[32], sb2[16], sW3[16];
  for (int i = threadIdx.x; i < 64 * 32; i += 256) sW1[i] = W1[i];
  for (int i = threadIdx.x; i < 32 * 16; i += 256) sW2[i] = W2[i];
  if (threadIdx.x < 32) sb1[threadIdx.x] = b1[threadIdx.x];
  if (threadIdx.x < 16) { sb2[threadIdx.x] = b2[threadIdx.x]; sW3[threadIdx.x] = W3[threadIdx.x]; }
  __syncthreads();
  int n = blockIdx.x * blockDim.x + threadIdx.x;
  if (n >= NN) return;
  float x[64];
#pragma unroll
  for (int c = 0; c < 64; ++c) x[c] = h[(size_t)n * 64 + c];
  float y1[32];
#pragma unroll
  for (int j = 0; j < 32; ++j) {
    float a = sb1[j];
    for (int c = 0; c < 64; ++c) a += x[c] * sW1[c * 32 + j];
    y1[j] = fmaxf(a, 0.f);
  }
  float y2[16];
#pragma unroll
  for (int j = 0; j < 16; ++j) {
    float a = sb2[j];
    for (int c = 0; c < 32; ++c) a += y1[c] * sW2[c * 16 + j];
    y2[j] = fmaxf(a, 0.f);
  }
  float a = b3[0];
#pragma unroll
  for (int c = 0; c < 16; ++c) a += y2[c] * sW3[c];
  out[n] = 1.f / (1.f + __expf(-a));
}

static inline int cdiv(long long a, int b) { return (int)((a + b - 1) / b); }

extern "C" void kernel_launch(void* const* d_in, const int* in_sizes, int n_in,
                              void* d_out, int out_size, void* d_ws, size_t ws_size,
                              hipStream_t stream) {
  // setup_inputs() dict order (insertion order, nested dicts flattened in order)
  const float* x    = (const float*)d_in[0];
  const int*   ei   = (const int*)d_in[1];
  const float* g1W  = (const float*)d_in[2];
  const float* g1as = (const float*)d_in[3];
  const float* g1ad = (const float*)d_in[4];
  const float* g1b  = (const float*)d_in[5];
  const float* bn1g = (const float*)d_in[6],  *bn1b = (const float*)d_in[7];
  const float* bn1m = (const float*)d_in[8],  *bn1v = (const float*)d_in[9];
  const float* c2W  = (const float*)d_in[10], *c2b  = (const float*)d_in[11];
  const float* bn2g = (const float*)d_in[12], *bn2b = (const float*)d_in[13];
  const float* bn2m = (const float*)d_in[14], *bn2v = (const float*)d_in[15];
  const float* g3W  = (const float*)d_in[16];
  const float* g3as = (const float*)d_in[17];
  const float* g3ad = (const float*)d_in[18];
  const float* g3b  = (const float*)d_in[19];
  const float* bn3g = (const float*)d_in[20], *bn3b = (const float*)d_in[21];
  const float* bn3m = (const float*)d_in[22], *bn3v = (const float*)d_in[23];
  const float* c4W  = (const float*)d_in[24], *c4b  = (const float*)d_in[25];
  const float* bn4g = (const float*)d_in[26], *bn4b = (const float*)d_in[27];
  const float* bn4m = (const float*)d_in[28], *bn4v = (const float*)d_in[29];
  const float* skW  = (const float*)d_in[30], *skb  = (const float*)d_in[31];
  const float* hW1  = (const float*)d_in[32], *hb1  = (const float*)d_in[33];
  const float* hW2  = (const float*)d_in[34], *hb2  = (const float*)d_in[35];
  const float* hW3  = (const float*)d_in[36], *hb3  = (const float*)d_in[37];
  float* out = (float*)d_out;

  // ---- workspace carve-up (floats) ----
  float* W = (float*)d_ws;
  size_t o = 0;
  float*    dinv  = W + o; o += NN;
  float*    hbuf  = W + o; o += (size_t)NN * 64;
  float*    skip  = W + o; o += (size_t)NN * 64;
  float*    hfeat = W + o; o += (size_t)NN * 256;
  float*    agg   = W + o; o += (size_t)NN * 256;
  float*    a_s   = W + o; o += (size_t)NN * 4;
  float*    a_d   = W + o; o += (size_t)NN * 4;
  unsigned* emax  = (unsigned*)(W + o); o += (size_t)NN * 4;
  float*    denom = W + o; o += (size_t)NN * 4;
  float*    alpha = W + o; o += (size_t)ET * 4;

  const int B = 256;

  // ---- degrees (self loop => init 1) and D^{-1/2} ----
  fill_f32<<<cdiv(NN, B), B, 0, stream>>>(dinv, 1.0f, NN);
  deg_accum<<<cdiv(NE, B), B, 0, stream>>>(ei, dinv);
  rsqrt_ip<<<cdiv(NN, B), B, 0, stream>>>(dinv, NN);

  // ---- skip = x @ Wsk + bsk  (K=7, Nc=64, one wave covers all 4 col tiles)
  wmma_gemm<7, 4><<<dim3(NN / 16, 1), 32, 0, stream>>>(x, skW, skb, skip, 64);

  // ---- GAT attention pipeline (shared) ----
  auto gat_edges = [&](const float* asrc_p, const float* adst_p, int Hh, int cols) {
    att_dots<<<cdiv((long long)NN * Hh, B), B, 0, stream>>>(hfeat, asrc_p, adst_p,
                                                            a_s, a_d, Hh);
    fill_ord<<<cdiv((long long)NN * Hh, B), B, 0, stream>>>(emax, -1e30f, NN * Hh);
    hipMemsetAsync(denom, 0, (size_t)NN * Hh * sizeof(float), stream);
    hipMemsetAsync(agg, 0, (size_t)NN * cols * sizeof(float), stream);
    gat_edge_max<<<cdiv((long long)ET * Hh, B), B, 0, stream>>>(ei, a_s, a_d, emax, Hh);
    gat_edge_expsum<<<cdiv((long long)ET * Hh, B), B, 0, stream>>>(ei, a_s, a_d,
                                                                   emax, denom, Hh);
    gat_alpha<<<cdiv((long long)ET * Hh, B), B, 0, stream>>>(ei, a_s, a_d, emax,
                                                             denom, alpha, Hh);
    gat_edge_agg<<<cdiv((long long)ET * cols, B), B, 0, stream>>>(ei, hfeat, alpha,
                                                                  agg, Hh, cols);
  };

  // ---- layer 1: GAT(7 -> 64, heads=4), residual = skip ----
  wmma_gemm<7, 4><<<dim3(NN / 16, 4), 32, 0, stream>>>(x, g1W, nullptr, hfeat, 256);
  gat_edges(g1as, g1ad, 4, 256);
  post_kernel<<<cdiv((long long)NN * 64, B), B, 0, stream>>>(agg, 4, g1b, bn1g, bn1b,
                                                             bn1m, bn1v, skip, hbuf);

  // ---- layer 2: GCN(64 -> 64), residual = h ----
  wmma_gemm<64, 4><<<dim3(NN / 16, 1), 32, 0, stream>>>(hbuf, c2W, nullptr, hfeat, 64);
  hipMemsetAsync(agg, 0, (size_t)NN * 64 * sizeof(float), stream);
  gcn_edge_agg<<<cdiv((long long)ET * 64, B), B, 0, stream>>>(ei, hfeat, dinv, agg);
  post_kernel<<<cdiv((long long)NN * 64, B), B, 0, stream>>>(agg, 1, c2b, bn2g, bn2b,
                                                             bn2m, bn2v, hbuf, hbuf);

  // ---- layer 3: GAT(64 -> 64, heads=2), residual = h ----
  wmma_gemm<64, 4><<<dim3(NN / 16, 2), 32, 0, stream>>>(hbuf, g3W, nullptr, hfeat, 128);
  gat_edges(g3as, g3ad, 2, 128);
  post_kernel<<<cdiv((long long)NN * 64, B), B, 0, stream>>>(agg, 2, g3b, bn3g, bn3b,
                                                             bn3m, bn3v, hbuf, hbuf);

  // ---- layer 4: GCN(64 -> 64), residual = h ----
  wmma_gemm<64, 4><<<dim3(NN / 16, 1), 32, 0, stream>>>(hbuf, c4W, nullptr, hfeat, 64);
  hipMemsetAsync(agg, 0, (size_t)NN * 64 * sizeof(float), stream);
  gcn_edge_agg<<<cdiv((long long)ET * 64, B), B, 0, stream>>>(ei, hfeat, dinv, agg);
  post_kernel<<<cdiv((long long)NN * 64, B), B, 0, stream>>>(agg, 1, c4b, bn4g, bn4b,
                                                             bn4m, bn4v, hbuf, hbuf);

  // ---- head MLP -> sigmoid -> out[N] ----
  head_kernel<<<cdiv(NN, 256), 256, 0, stream>>>(hbuf, hW1, hb1, hW2, hb2, hW3, hb3, out);

  (void)in_sizes; (void)n_in; (void)out_size; (void)ws_size;
}